// LogisticRegressionModel_31421980737603
// MI455X (gfx1250) — compile-verified
//
#include <hip/hip_runtime.h>
#include <math.h>

typedef __attribute__((ext_vector_type(2))) float v2f;
typedef __attribute__((ext_vector_type(8))) float v8f;

#define NDIM    16
#define BATCHN  32768
#define NPREFIX 969   // 1 + 16 + 136 + 816 prefix features
#define KPAD    972   // NPREFIX padded to multiple of 4 (WMMA K step)
#define BROWS   976   // B rows allocated & zeroed (safety margin)

// d_ws layout:
//   [0, BROWS*16*4)          B matrix  float [BROWS][16]
//   [65536, 65536+969*8)     parent table int2 {prefixIdx, lastIdx}

__global__ void poly_zero(float* __restrict__ B, int n) {
    int i = blockIdx.x * blockDim.x + threadIdx.x;
    if (i < n) B[i] = 0.0f;
}

// Single-thread setup: enumerate monomials in exact combinations_with_replacement
// order, building B[p][n] = w_deg * W[monomial] and the prefix parent table.
__global__ void poly_setup(const float* __restrict__ W,
                           const float* __restrict__ Mraw,
                           float* __restrict__ B,
                           int2* __restrict__ par) {
    if (threadIdx.x != 0 || blockIdx.x != 0) return;
    float M = 1.0f + 3.0f / (1.0f + expf(-Mraw[0]));   // sigmoid(M_raw)*(4-1)+1
    float wd[5];
    for (int d = 1; d <= 4; ++d)
        wd[d] = 1.0f / (1.0f + expf(-10.0f * (M - (float)d + 0.5f)));

    int widx = 1;                       // W[0] is the constant feature, added later
    // degree 1: prefix = const (p=0), last = n
    for (int n = 0; n < NDIM; ++n) B[0 * 16 + n] = wd[1] * W[widx++];
    // degree 2: prefix = deg1 monomial i (p = 1+i), last = n >= i
    for (int i = 0; i < NDIM; ++i)
        for (int n = i; n < NDIM; ++n) B[(1 + i) * 16 + n] = wd[2] * W[widx++];
    // degree 3: prefix = deg2 monomial (i,j) (p = q), last = n >= j
    int q = 17;
    for (int i = 0; i < NDIM; ++i)
        for (int j = i; j < NDIM; ++j) {
            par[q] = make_int2(1 + i, j);          // Pf[q] = Pf[1+i] * x[j]
            for (int n = j; n < NDIM; ++n) B[q * 16 + n] = wd[3] * W[widx++];
            ++q;
        }
    // degree 4: prefix = deg3 monomial (i,j,k) (p = t), last = n >= k
    int t = 153;
    q = 17;
    for (int i = 0; i < NDIM; ++i)
        for (int j = i; j < NDIM; ++j) {
            for (int k = j; k < NDIM; ++k) {
                par[t] = make_int2(q, k);          // Pf[t] = Pf[q] * x[k]
                for (int n = k; n < NDIM; ++n) B[t * 16 + n] = wd[4] * W[widx++];
                ++t;
            }
            ++q;
        }
}

// One wave32 per 16-sample tile. Builds 16 x KPAD prefix features in LDS,
// contracts against B with V_WMMA_F32_16X16X4_F32, then row-reduces x*C.
__global__ __launch_bounds__(32)
void poly_main(const float* __restrict__ x,
               const float* __restrict__ W,
               const float* __restrict__ bvec,
               const float* __restrict__ B,
               const int2* __restrict__ par,
               float* __restrict__ out) {
    __shared__ float Pf[16][KPAD];   // prefix feature tile (~62 KB)
    __shared__ float xt[16][16];
    __shared__ float red[16][16];

    const int lane = threadIdx.x;
    const int s0   = blockIdx.x * 16;

    // load x tile (16 samples x 16 dims, row-major)
    for (int e = lane; e < 256; e += 32) xt[e >> 4][e & 15] = x[s0 * NDIM + e];
    __syncthreads();

    // const + degree-1 prefixes
    for (int e = lane; e < 16 * 17; e += 32) {
        int m = e / 17, c = e % 17;
        Pf[m][c] = (c == 0) ? 1.0f : xt[m][c - 1];
    }
    // zero K padding
    for (int e = lane; e < 16 * (KPAD - NPREFIX); e += 32) {
        int m = e / (KPAD - NPREFIX), c = e % (KPAD - NPREFIX);
        Pf[m][NPREFIX + c] = 0.0f;
    }
    __syncthreads();
    // degree-2 prefixes p in [17,152]
    for (int e = lane; e < 136 * 16; e += 32) {
        int f = e >> 4, m = e & 15, p = 17 + f;
        int2 pr = par[p];
        Pf[m][p] = Pf[m][pr.x] * xt[m][pr.y];
    }
    __syncthreads();
    // degree-3 prefixes p in [153,968]
    for (int e = lane; e < 816 * 16; e += 32) {
        int f = e >> 4, m = e & 15, p = 153 + f;
        int2 pr = par[p];
        Pf[m][p] = Pf[m][pr.x] * xt[m][pr.y];
    }
    __syncthreads();

    // C[16x16] = sum_k Pf[m,k] * B[k,n] via f32 WMMA, K in steps of 4
    v8f acc = {};
    const int row   = lane & 15;            // A: M row
    const int col   = lane & 15;            // B: N column
    const int khalf = (lane >> 4) << 1;     // lanes 16-31 hold K+2,K+3
    for (int k0 = 0; k0 < KPAD; k0 += 4) {
        v2f a;
        a.x = Pf[row][k0 + khalf];
        a.y = Pf[row][k0 + khalf + 1];
        v2f bb;
        bb.x = B[(k0 + khalf) * 16 + col];
        bb.y = B[(k0 + khalf + 1) * 16 + col];
        acc = __builtin_amdgcn_wmma_f32_16x16x4_f32(
            false, a, false, bb, (short)0, acc, false, false);
    }

    // combine: logit[m] = b + W[0] + sum_n x[m,n] * C[m,n]
    const int mbase = (lane >> 4) << 3;     // C layout: lane>=16 holds rows 8..15
#pragma unroll
    for (int v = 0; v < 8; ++v)
        red[mbase + v][col] = acc[v] * xt[mbase + v][col];
    __syncthreads();
    if (lane < 16) {
        float s = 0.0f;
#pragma unroll
        for (int n = 0; n < 16; ++n) s += red[lane][n];
        float logit = s + bvec[0] + W[0];
        out[s0 + lane] = 1.0f / (1.0f + expf(-logit));
    }
}

extern "C" void kernel_launch(void* const* d_in, const int* in_sizes, int n_in,
                              void* d_out, int out_size, void* d_ws, size_t ws_size,
                              hipStream_t stream) {
    const float* x    = (const float*)d_in[0];
    const float* W    = (const float*)d_in[1];
    const float* b    = (const float*)d_in[2];
    const float* Mraw = (const float*)d_in[3];

    float* B   = (float*)d_ws;
    int2*  par = (int2*)((char*)d_ws + 65536);
    float* out = (float*)d_out;

    const int nB = BROWS * 16;
    poly_zero<<<(nB + 255) / 256, 256, 0, stream>>>(B, nB);
    poly_setup<<<1, 1, 0, stream>>>(W, Mraw, B, par);
    poly_main<<<BATCHN / 16, 32, 0, stream>>>(x, W, b, B, par, out);
}